// PLLBank_53111565582369
// MI455X (gfx1250) — compile-verified
//
#include <hip/hip_runtime.h>
#include <cmath>
#include <complex>

#ifndef M_PI
#define M_PI 3.14159265358979323846
#endif

#define T_LEN 32000
#define NSEQ  256          // BS*C
#define LH_   2048
#define NCH   64
#define DDEC  160
#define NDEC  200
#define PAD   15
#define EXT_LEN (T_LEN + 2*PAD)

typedef __attribute__((ext_vector_type(2))) float v2f;
typedef __attribute__((ext_vector_type(8))) float v8f;

struct Coef { float b0,b1,b2,b3,b4,a1,a2,a3,a4,z0,z1,z2,z3; };

// ---------------------------------------------------------------------------
// Phase 1: filterbank convolution as GEMM via V_WMMA_F32_16X16X4_F32.
// Block = 256 thr (8 waves). Block covers 32 time rows x 64 channels for one
// batch; each wave owns one 16(time) x 16(chan) f32 accumulator tile.
// out_t layout: [t][seq] with seq = b*64 + c  (coalesced for later scans).
// ---------------------------------------------------------------------------
__global__ __launch_bounds__(256) void pll_conv_wmma(
    const float* __restrict__ wav,      // (BS, T)
    const float* __restrict__ ker,      // (LH, C) row-major
    float* __restrict__ out_t)          // (T, 256)
{
  __shared__ float s_wav[2080];         // wav window [t0-1023, t0+1056]
  __shared__ float s_ker[128 * 64];     // K-chunk of the kernel matrix

  const int b    = blockIdx.y;
  const int t0   = blockIdx.x * 32;
  const int tid  = threadIdx.x;
  const int lane = tid & 31;
  const int wid  = tid >> 5;
  const int tt   = wid >> 2;            // 0..1 time tile within block
  const int c0   = (wid & 3) * 16;      // channel tile base
  const int m    = lane & 15;           // A row / B col / D col
  const int kh   = lane >> 4;           // K half-select (ISA f32 A layout)
  const int wbase = t0 - 1023;

  for (int i = tid; i < 2080; i += 256) {
    int j = wbase + i;
    s_wav[i] = (j >= 0 && j < T_LEN) ? wav[b * T_LEN + j] : 0.0f;
  }

  v8f acc = {0.f,0.f,0.f,0.f,0.f,0.f,0.f,0.f};
  const int tw = t0 + tt * 16;

  for (int kc = 0; kc < LH_; kc += 128) {
    __syncthreads();
    for (int i = tid; i < 128 * 64; i += 256) s_ker[i] = ker[kc * 64 + i];
    __syncthreads();
    if (kc + 128 < LH_)
      __builtin_prefetch(&ker[(kc + 128) * 64 + tid * 8], 0, 1);

    for (int ks = 0; ks < 128; ks += 4) {
      // A[m][k] = wav[tw + m + 1024 - k]; this lane holds K = klo, klo+1
      const int klo = kc + ks + 2 * kh;
      const int x   = tw + m + 1024 - klo - wbase;
      v2f a;  a.x = s_wav[x];      a.y = s_wav[x - 1];
      // B[k][n] = ker[k][c0+n]; this lane: N = m, K = kb, kb+1
      const int kb = ks + 2 * kh;
      v2f bm; bm.x = s_ker[kb * 64 + c0 + m];
              bm.y = s_ker[(kb + 1) * 64 + c0 + m];
      acc = __builtin_amdgcn_wmma_f32_16x16x4_f32(
                false, a, false, bm, (short)0, acc, false, false);
    }
  }

  // D layout: VGPR r -> M = 8*kh + r, N = lane&15
  #pragma unroll
  for (int r = 0; r < 8; ++r) {
    int t = tw + kh * 8 + r;
    out_t[(size_t)t * NSEQ + b * NCH + c0 + m] = acc[r];
  }
}

// ---------------------------------------------------------------------------
// Phase 2: PLL scan. One lane per (b,c) sequence; 8 wave32 blocks.
// Writes lock, freq, and zero-cross indicator streams in [t][seq] layout.
// ---------------------------------------------------------------------------
__global__ __launch_bounds__(32) void pll_scan(
    const float* __restrict__ out_t, const float* __restrict__ fp,
    float* __restrict__ lockb, float* __restrict__ freqb,
    float* __restrict__ crossb,
    float g1, float g2, float blk0, float blk1, float alk1)
{
  const int seq = blockIdx.x * 32 + threadIdx.x;
  const int c   = seq & (NCH - 1);
  const float wo = 6.283185307179586f * fp[c] * (1.0f / 16000.0f);
  const float FS_O2PI = 16000.0f * 0.15915494309189535f;   // Fs/(2*pi)
  float theta = 0.f, integ = 0.f, zl = 0.f, vprev = 0.f;

  for (int t = 0; t < T_LEN; ++t) {
    float xt = out_t[(size_t)t * NSEQ + seq];
    float s, cv;
    __sincosf(theta, &s, &cv);
    float e   = -xt * s * 10.0f;               // AGC = 10
    integ    += g2 * e;
    float dth = wo + g1 * e + integ;
    theta    += dth;
    float lr  = xt * cv;
    float y   = blk0 * lr + zl;
    zl        = blk1 * lr - alk1 * y;
    freqb[(size_t)t * NSEQ + seq] = dth * FS_O2PI;
    lockb[(size_t)t * NSEQ + seq] = y;
    if (t > 0)
      crossb[(size_t)(t - 1) * NSEQ + seq] =
          (vprev > 0.0f && !(cv > 0.0f)) ? 1.0f : 0.0f;
    vprev = cv;
  }
  crossb[(size_t)(T_LEN - 1) * NSEQ + seq] = 0.0f;   // aux last element = 0
}

// ---------------------------------------------------------------------------
// Phase 3: filtfilt (order-4 Butterworth, DF2-transposed), freq+lock fused.
// Writes only decimated samples of the backward pass.
// ---------------------------------------------------------------------------
__device__ __forceinline__ float ext_at(const float* x, int seq, int i) {
  if (i < PAD) return 2.0f * x[seq] - x[(size_t)(PAD - i) * NSEQ + seq];
  int t = i - PAD;
  if (t < T_LEN) return x[(size_t)t * NSEQ + seq];
  int k = t - T_LEN;   // 0..14
  return 2.0f * x[(size_t)(T_LEN - 1) * NSEQ + seq]
             - x[(size_t)(T_LEN - 2 - k) * NSEQ + seq];
}

__global__ __launch_bounds__(32) void pll_filtfilt(
    const float* __restrict__ freqb, const float* __restrict__ lockb,
    float* __restrict__ yscr0, float* __restrict__ yscr1,
    float* __restrict__ freqd, float* __restrict__ lockd,
    Coef cf, Coef cl)
{
  const int id  = blockIdx.x * 32 + threadIdx.x;   // 0..511
  const int sig = id >> 8;                          // 0 = freq, 1 = lock
  const int seq = id & (NSEQ - 1);
  const float* x = sig ? lockb : freqb;
  float* yscr    = sig ? yscr1 : yscr0;
  float* dec     = sig ? lockd : freqd;
  const Coef C   = sig ? cl : cf;

  float x0 = ext_at(x, seq, 0);
  float z0 = C.z0 * x0, z1 = C.z1 * x0, z2 = C.z2 * x0, z3 = C.z3 * x0;
  for (int i = 0; i < EXT_LEN; ++i) {
    float xt = ext_at(x, seq, i);
    float y  = C.b0 * xt + z0;
    z0 = C.b1 * xt + z1 - C.a1 * y;
    z1 = C.b2 * xt + z2 - C.a2 * y;
    z2 = C.b3 * xt + z3 - C.a3 * y;
    z3 = C.b4 * xt       - C.a4 * y;
    yscr[(size_t)i * NSEQ + seq] = y;
  }
  float yl = yscr[(size_t)(EXT_LEN - 1) * NSEQ + seq];
  z0 = C.z0 * yl; z1 = C.z1 * yl; z2 = C.z2 * yl; z3 = C.z3 * yl;
  for (int i = 0; i < EXT_LEN; ++i) {
    float xt = yscr[(size_t)(EXT_LEN - 1 - i) * NSEQ + seq];
    float y  = C.b0 * xt + z0;
    z0 = C.b1 * xt + z1 - C.a1 * y;
    z1 = C.b2 * xt + z2 - C.a2 * y;
    z2 = C.b3 * xt + z3 - C.a3 * y;
    z3 = C.b4 * xt       - C.a4 * y;
    int t = T_LEN + 14 - i;          // final (un-reversed, un-padded) index
    if (t >= 0 && t < T_LEN && (t % DDEC) == 0)
      dec[(size_t)(t / DDEC) * NSEQ + seq] = y;
  }
}

// ---------------------------------------------------------------------------
// Phase 4: triangular smoothing of zero-crossings at decimated points only,
// dz across channels via LDS, final transpose + fp passthrough.
// ---------------------------------------------------------------------------
__global__ __launch_bounds__(256) void pll_finalize(
    const float* __restrict__ crossb, const float* __restrict__ lockd,
    const float* __restrict__ freqd,  const float* __restrict__ fp,
    float* __restrict__ outp)
{
  __shared__ float s_zc[NSEQ];
  const int d   = blockIdx.x;        // decimated index 0..199
  const int seq = threadIdx.x;
  const int b   = seq >> 6, c = seq & (NCH - 1);
  const int t   = d * DDEC;
  int mlo = t - 399; if (mlo < 0) mlo = 0;
  int mhi = t + 399; if (mhi > T_LEN - 1) mhi = T_LEN - 1;
  float acc = 0.0f;
  for (int mm = mlo; mm <= mhi; ++mm) {
    int dd = t - mm; if (dd < 0) dd = -dd;
    acc += (float)(400 - dd) * crossb[(size_t)mm * NSEQ + seq];
  }
  s_zc[seq] = acc;
  __syncthreads();
  int cc = (c < 1) ? 1 : c;
  float dz = (s_zc[b * NCH + cc] - s_zc[b * NCH + cc - 1]) / fp[c];
  size_t obase = ((((size_t)b * NDEC + d) * NCH) + c) * 4;
  outp[obase + 0] = lockd[(size_t)d * NSEQ + seq];
  outp[obase + 1] = freqd[(size_t)d * NSEQ + seq];
  outp[obase + 2] = acc;
  outp[obase + 3] = dz;
  if (d == 0 && seq < NCH) outp[(size_t)4 * NDEC * NCH * 4 + seq] = fp[seq];
}

// ---------------------------------------------------------------------------
// Host helpers: Butterworth design + lfilter_zi (double precision, per call).
// ---------------------------------------------------------------------------
static void butter_lp4(double Wn, double b[5], double a[5]) {
  const int N = 4;
  const double fs = 2.0;
  double warped = 2.0 * fs * std::tan(M_PI * Wn / fs);
  std::complex<double> p[4], pz[4], prod(1.0, 0.0);
  for (int k = 1; k <= N; ++k)
    p[k-1] = warped * std::exp(std::complex<double>(0.0, M_PI * (2.0*k + N - 1.0) / (2.0*N)));
  double fs2 = 2.0 * fs;
  for (int i = 0; i < 4; ++i) { pz[i] = (fs2 + p[i]) / (fs2 - p[i]); prod *= (fs2 - p[i]); }
  double gain = std::pow(warped, (double)N) / prod.real();
  const double binom[5] = {1, 4, 6, 4, 1};
  for (int i = 0; i < 5; ++i) b[i] = gain * binom[i];
  std::complex<double> coef[5] = {1, 0, 0, 0, 0};
  for (int i = 0; i < 4; ++i)
    for (int j = 4; j >= 1; --j) coef[j] -= pz[i] * coef[j-1];
  for (int i = 0; i < 5; ++i) a[i] = coef[i].real();
}

static void lfilter_zi4(const double b_[5], const double a_[5], double zi[4]) {
  double a[5], b[5];
  for (int i = 0; i < 5; ++i) { a[i] = a_[i] / a_[0]; b[i] = b_[i] / a_[0]; }
  double comp[4][4] = {};
  for (int j = 0; j < 4; ++j) comp[0][j] = -a[j+1];
  comp[1][0] = 1; comp[2][1] = 1; comp[3][2] = 1;
  double M[4][5];
  for (int i = 0; i < 4; ++i) {
    for (int j = 0; j < 4; ++j) M[i][j] = (i == j ? 1.0 : 0.0) - comp[j][i];
    M[i][4] = b[i+1] - a[i+1] * b[0];
  }
  for (int col = 0; col < 4; ++col) {
    int piv = col;
    for (int r = col + 1; r < 4; ++r)
      if (std::fabs(M[r][col]) > std::fabs(M[piv][col])) piv = r;
    if (piv != col)
      for (int k = 0; k < 5; ++k) { double t = M[col][k]; M[col][k] = M[piv][k]; M[piv][k] = t; }
    for (int r = 0; r < 4; ++r) {
      if (r == col) continue;
      double f = M[r][col] / M[col][col];
      for (int k = col; k < 5; ++k) M[r][k] -= f * M[col][k];
    }
  }
  for (int i = 0; i < 4; ++i) zi[i] = M[i][4] / M[i][i];
}

static Coef make_coef(const double b[5], const double a[5], const double zi[4]) {
  Coef c;
  c.b0 = (float)b[0]; c.b1 = (float)b[1]; c.b2 = (float)b[2]; c.b3 = (float)b[3]; c.b4 = (float)b[4];
  c.a1 = (float)a[1]; c.a2 = (float)a[2]; c.a3 = (float)a[3]; c.a4 = (float)a[4];
  c.z0 = (float)zi[0]; c.z1 = (float)zi[1]; c.z2 = (float)zi[2]; c.z3 = (float)zi[3];
  return c;
}

extern "C" void kernel_launch(void* const* d_in, const int* in_sizes, int n_in,
                              void* d_out, int out_size, void* d_ws, size_t ws_size,
                              hipStream_t stream) {
  (void)in_sizes; (void)n_in; (void)out_size; (void)ws_size;
  const float* wav = (const float*)d_in[0];   // (4, 32000)
  const float* ker = (const float*)d_in[1];   // (2048, 64)
  const float* fp  = (const float*)d_in[2];   // (64,)
  float* outp = (float*)d_out;

  // Workspace layout (f32), ~164 MB total.
  float* ws = (float*)d_ws;
  size_t off = 0;
  float* out_t  = ws + off; off += (size_t)(T_LEN + 32) * NSEQ;  // also yscr0
  float* lockb  = ws + off; off += (size_t)T_LEN * NSEQ;
  float* freqb  = ws + off; off += (size_t)T_LEN * NSEQ;
  float* crossb = ws + off; off += (size_t)T_LEN * NSEQ;
  float* yscr1  = ws + off; off += (size_t)(T_LEN + 32) * NSEQ;
  float* freqd  = ws + off; off += (size_t)NDEC * NSEQ;
  float* lockd  = ws + off; off += (size_t)NDEC * NSEQ;
  float* yscr0  = out_t;    // out_t is dead after pll_scan

  // Filter designs (match reference's scipy-free butter + lfilter_zi).
  double bf[5], af[5], zf[4], bl[5], al[5], zl[4];
  butter_lp4(25.0  * 2.0 / 16000.0, bf, af); lfilter_zi4(bf, af, zf);
  butter_lp4(100.0 * 2.0 / 16000.0, bl, al); lfilter_zi4(bl, al, zl);
  Coef cf = make_coef(bf, af, zf);
  Coef cl = make_coef(bl, al, zl);

  // PLL constants (FN=200, SITA=0.7, FC_LOCK=4).
  const double wn = 2.0 * M_PI * 200.0 / 16000.0;
  const double sita = 0.7;
  const float g1 = (float)(1.0 - std::exp(-2.0 * sita * wn));
  const float g2 = (float)(1.0 + std::exp(-2.0 * sita * wn)
                   - 2.0 * std::exp(-sita * wn) * std::cos(wn * std::sqrt(1.0 - sita*sita)));
  const double wc = 2.0 * M_PI * 4.0, c2 = 2.0 * 16000.0;
  const float blk0 = (float)(wc / (c2 + wc));
  const float blk1 = blk0;
  const float alk1 = (float)((wc - c2) / (c2 + wc));

  dim3 gConv(T_LEN / 32, 4);
  pll_conv_wmma<<<gConv, 256, 0, stream>>>(wav, ker, out_t);
  pll_scan<<<8, 32, 0, stream>>>(out_t, fp, lockb, freqb, crossb,
                                 g1, g2, blk0, blk1, alk1);
  pll_filtfilt<<<16, 32, 0, stream>>>(freqb, lockb, yscr0, yscr1,
                                      freqd, lockd, cf, cl);
  pll_finalize<<<NDEC, 256, 0, stream>>>(crossb, lockd, freqd, fp, outp);
}